// LIGHTGCN_29918742184782
// MI455X (gfx1250) — compile-verified
//
#include <hip/hip_runtime.h>
#include <hip/hip_bf16.h>
#include <math.h>

#define N_USERS 100000
#define N_ITEMS 50000
#define N_NODES (N_USERS + N_ITEMS)
#define EMB_DIM 64
#define N_EDGES 6400000
#define N_PAIRS 16384

typedef __attribute__((ext_vector_type(2))) float v2f;
typedef __attribute__((ext_vector_type(8))) float v8f;

// ---------------------------------------------------------------------------
// Kernel 1: concatenate user/item embeddings into emb0 and initialize acc.
// Element space: N_NODES*EMB_DIM floats, processed as float4.
// ---------------------------------------------------------------------------
__global__ void lg_init_kernel(const float* __restrict__ user_emb,
                               const float* __restrict__ item_emb,
                               float* __restrict__ emb,
                               float* __restrict__ acc) {
    const size_t USER_V4 = (size_t)N_USERS * EMB_DIM / 4;
    const size_t TOTAL_V4 = (size_t)N_NODES * EMB_DIM / 4;
    size_t i = (size_t)blockIdx.x * blockDim.x + threadIdx.x;
    if (i >= TOTAL_V4) return;
    float4 v;
    if (i < USER_V4) {
        v = ((const float4*)user_emb)[i];
    } else {
        v = ((const float4*)item_emb)[i - USER_V4];
    }
    ((float4*)emb)[i] = v;
    ((float4*)acc)[i] = v;
}

// ---------------------------------------------------------------------------
// Kernel 2: zero a node-embedding buffer (float4 granularity).
// ---------------------------------------------------------------------------
__global__ void lg_zero_kernel(float* __restrict__ buf) {
    const size_t TOTAL_V4 = (size_t)N_NODES * EMB_DIM / 4;
    size_t i = (size_t)blockIdx.x * blockDim.x + threadIdx.x;
    if (i >= TOTAL_V4) return;
    ((float4*)buf)[i] = make_float4(0.f, 0.f, 0.f, 0.f);
}

// ---------------------------------------------------------------------------
// Kernel 3: edge scatter — dst[row] += val * src[col], 16 lanes per edge,
// each lane handles 4 consecutive dims (coalesced 256B gather per edge).
// Source table (38.4MB) is L2-resident; atomics are non-returning f32 adds
// serviced by L2 atomic units. Edge stream prefetched with global_prefetch_b8.
// ---------------------------------------------------------------------------
__global__ void lg_scatter_kernel(const int* __restrict__ adj_row,
                                  const int* __restrict__ adj_col,
                                  const float* __restrict__ adj_val,
                                  const float* __restrict__ src,
                                  float* __restrict__ dst) {
    int t = blockIdx.x * blockDim.x + threadIdx.x;
    int e = t >> 4;              // 16 lanes per edge
    if (e >= N_EDGES) return;
    int l = t & 15;              // lane-in-edge: dims [4l, 4l+3]

    if (l == 0) {
        // Speculative prefetch of the edge stream ahead of us (dropped if OOB).
        __builtin_prefetch(adj_row + e + 4096, 0, 1);
        __builtin_prefetch(adj_col + e + 4096, 0, 1);
        __builtin_prefetch(adj_val + e + 4096, 0, 1);
    }

    int r = adj_row[e];
    int c = adj_col[e];
    float w = adj_val[e];

    const float4 s = *(const float4*)(src + (size_t)c * EMB_DIM + l * 4);
    float* d = dst + (size_t)r * EMB_DIM + l * 4;

    unsafeAtomicAdd(d + 0, w * s.x);   // global_atomic_add_f32 (no return)
    unsafeAtomicAdd(d + 1, w * s.y);
    unsafeAtomicAdd(d + 2, w * s.z);
    unsafeAtomicAdd(d + 3, w * s.w);
}

// ---------------------------------------------------------------------------
// Kernel 4: acc += layer output (float4 granularity).
// ---------------------------------------------------------------------------
__global__ void lg_accum_kernel(float* __restrict__ acc,
                                const float* __restrict__ layer) {
    const size_t TOTAL_V4 = (size_t)N_NODES * EMB_DIM / 4;
    size_t i = (size_t)blockIdx.x * blockDim.x + threadIdx.x;
    if (i >= TOTAL_V4) return;
    float4 a = ((float4*)acc)[i];
    float4 b = ((const float4*)layer)[i];
    a.x += b.x; a.y += b.y; a.z += b.z; a.w += b.w;
    ((float4*)acc)[i] = a;
}

// ---------------------------------------------------------------------------
// Kernel 5: scoring via V_WMMA_F32_16X16X4_F32.
// One wave32 handles 16 (user,item) pairs. A = 16 user rows (16xK), B^T = 16
// item rows (so B[k][n] = item_n[k]); D = A x B gives all 256 cross dots;
// the diagonal is the wanted per-pair dot. K=64 accumulated as 16 steps of 4.
//
// f32 A-operand layout (16x4, 2 VGPRs): lanes 0-15 hold M=lane, K=(0,1);
// lanes 16-31 hold M=lane-16, K=(2,3)  -> float2 load at k = 4t + 2*(lane>>4)
// from row-major user row. B (4x16) mirrors: lane holds N=lane&15, same K pair
// -> identical float2 load from the item row. Final = acc/4 for both sides, so
// scale dot by 1/16 before sigmoid.
// ---------------------------------------------------------------------------
__global__ void lg_score_kernel(const float* __restrict__ acc,
                                const int* __restrict__ user_ids,
                                const int* __restrict__ item_ids,
                                float* __restrict__ out) {
    __shared__ float tile[16][17];

    int lane = threadIdx.x;      // 0..31, single wave
    int l16 = lane & 15;
    int sub = lane >> 4;
    int base = blockIdx.x * 16;

    int uid = user_ids[base + l16];
    int iid = item_ids[base + l16];
    const float* urow = acc + (size_t)uid * EMB_DIM;
    const float* vrow = acc + ((size_t)N_USERS + (size_t)iid) * EMB_DIM;

    v8f c = {};
#pragma unroll
    for (int t = 0; t < 16; ++t) {
        int k = 4 * t + 2 * sub;
        float2 a2 = *(const float2*)(urow + k);
        float2 b2 = *(const float2*)(vrow + k);
        v2f a; a.x = a2.x; a.y = a2.y;
        v2f b; b.x = b2.x; b.y = b2.y;
        c = __builtin_amdgcn_wmma_f32_16x16x4_f32(
                /*neg_a=*/false, a, /*neg_b=*/false, b,
                /*c_mod=*/(short)0, c, /*reuse_a=*/false, /*reuse_b=*/false);
    }

    // D layout: VGPR r holds M = r + 8*sub (row), N = l16 (col).
#pragma unroll
    for (int r = 0; r < 8; ++r) {
        tile[r + 8 * sub][l16] = c[r];
    }
    __syncthreads();

    if (lane < 16) {
        float dot = tile[lane][lane] * (1.0f / 16.0f);  // (acc/4)·(acc/4)
        out[base + lane] = 1.0f / (1.0f + __expf(-dot));
    }
}

// ---------------------------------------------------------------------------
// Host-side launcher. Inputs (setup_inputs order):
//   0 user_emb f32[100000*64], 1 item_emb f32[50000*64],
//   2 adj_row i32[6.4M], 3 adj_col i32[6.4M], 4 adj_val f32[6.4M],
//   5 user_ids i32[16384], 6 item_ids i32[16384]
// Output: f32[16384] sigmoid scores.
// Workspace: acc | embA | embB, each N_NODES*64 f32 (3 x 38.4 MB).
// ---------------------------------------------------------------------------
extern "C" void kernel_launch(void* const* d_in, const int* in_sizes, int n_in,
                              void* d_out, int out_size, void* d_ws, size_t ws_size,
                              hipStream_t stream) {
    const float* user_emb = (const float*)d_in[0];
    const float* item_emb = (const float*)d_in[1];
    const int*   adj_row  = (const int*)d_in[2];
    const int*   adj_col  = (const int*)d_in[3];
    const float* adj_val  = (const float*)d_in[4];
    const int*   user_ids = (const int*)d_in[5];
    const int*   item_ids = (const int*)d_in[6];
    float* out = (float*)d_out;

    const size_t NODE_ELEMS = (size_t)N_NODES * EMB_DIM;
    float* acc  = (float*)d_ws;
    float* embA = acc + NODE_ELEMS;
    float* embB = embA + NODE_ELEMS;

    const int BLK = 256;
    const size_t TOTAL_V4 = NODE_ELEMS / 4;
    const int gridElem = (int)((TOTAL_V4 + BLK - 1) / BLK);

    lg_init_kernel<<<gridElem, BLK, 0, stream>>>(user_emb, item_emb, embA, acc);

    // 16 threads/edge, 256 threads/block -> 16 edges/block
    const int gridScatter = (N_EDGES * 16) / BLK;   // 400000 exactly

    float* cur = embA;
    float* nxt = embB;
    for (int layer = 0; layer < 3; ++layer) {
        lg_zero_kernel<<<gridElem, BLK, 0, stream>>>(nxt);
        lg_scatter_kernel<<<gridScatter, BLK, 0, stream>>>(adj_row, adj_col,
                                                           adj_val, cur, nxt);
        lg_accum_kernel<<<gridElem, BLK, 0, stream>>>(acc, nxt);
        float* tmp = cur; cur = nxt; nxt = tmp;
    }

    // 1024 blocks x 1 wave, 16 pairs per wave.
    lg_score_kernel<<<N_PAIRS / 16, 32, 0, stream>>>(acc, user_ids, item_ids, out);
}